// VectorQuantizer_59665685676278
// MI455X (gfx1250) — compile-verified
//
#include <hip/hip_runtime.h>
#include <hip/hip_bf16.h>

typedef __attribute__((ext_vector_type(2))) float v2f;
typedef __attribute__((ext_vector_type(8))) float v8f;

#define N_ROWS   131072          // 32*64*64
#define DDIM     64
#define KCB      512
#define E_STRIDE 68              // padded LDS row stride (dwords): conflict-free ds_load_b64
#define NTOT     (N_ROWS * DDIM) // 8388608 elements of qtised

// ---------------------------------------------------------------------------
// Kernel 1: per-row argmin over the 512 codewords using V_WMMA_F32_16X16X4_F32.
// dist proxy = 0.5*|e_k|^2 - <f, e_k>  (row-constant |f|^2 dropped; argmin-invariant)
// Two independent accumulator chains (2 column tiles / iter) for XDL pipelining.
// ---------------------------------------------------------------------------
__global__ __launch_bounds__(256) void vq_argmin_kernel(const float* __restrict__ X,
                                                        const float* __restrict__ E,
                                                        int* __restrict__ encIdx) {
  extern __shared__ float smem[];
  float* Elds = smem;                   // transposed codebook: Elds[n*E_STRIDE + d] = E[d*512 + n]
  float* hn   = smem + KCB * E_STRIDE;  // hn[n] = 0.5 * |e_n|^2

  const int tid = threadIdx.x;

  // Stage codebook transposed into LDS (coalesced global reads).
  for (int i = tid; i < DDIM * KCB; i += 256) {
    int d = i >> 9;          // row in E  (D dim)
    int n = i & (KCB - 1);   // col in E  (K dim)
    Elds[n * E_STRIDE + d] = E[i];
  }
  __syncthreads();

  // Precompute half squared norms.
  for (int n = tid; n < KCB; n += 256) {
    float s = 0.f;
    #pragma unroll 8
    for (int d = 0; d < DDIM; ++d) { float v = Elds[n * E_STRIDE + d]; s += v * v; }
    hn[n] = 0.5f * s;
  }
  __syncthreads();

  const int wave = tid >> 5;
  const int lane = tid & 31;
  const int half = lane >> 4;   // 0: rows M=0..7 / K=0,1   1: rows M=8..15 / K=2,3
  const int lr   = lane & 15;
  const int r0   = blockIdx.x * 128 + wave * 16;   // 8 waves x 16 rows

  // A fragments: 16x4 f32 tile layout — lanes 0-15 hold K={4kc,4kc+1}, lanes 16-31 K={4kc+2,4kc+3}
  v2f a[16];
  const float* xrow = X + (size_t)(r0 + lr) * DDIM + half * 2;
  #pragma unroll
  for (int kc = 0; kc < 16; ++kc) a[kc] = *(const v2f*)(xrow + kc * 4);

  float best[8];
  int   bidx[8];
  #pragma unroll
  for (int r = 0; r < 8; ++r) { best[r] = 3.4e38f; bidx[r] = 0; }

  for (int ct = 0; ct < 32; ct += 2) {       // 2 column tiles (32 codewords) per trip
    const int n0 = ct * 16;
    const float* ep0 = &Elds[(n0 + lr) * E_STRIDE + half * 2];
    const float* ep1 = ep0 + 16 * E_STRIDE;
    v8f c0 = {0.f, 0.f, 0.f, 0.f, 0.f, 0.f, 0.f, 0.f};
    v8f c1 = {0.f, 0.f, 0.f, 0.f, 0.f, 0.f, 0.f, 0.f};
    #pragma unroll
    for (int kc = 0; kc < 16; ++kc) {
      v2f b0 = *(const v2f*)(ep0 + kc * 4);  // B 4x16: lane holds column, 2 K values
      v2f b1 = *(const v2f*)(ep1 + kc * 4);
      c0 = __builtin_amdgcn_wmma_f32_16x16x4_f32(
               false, a[kc], false, b0, (short)0, c0, false, false);
      c1 = __builtin_amdgcn_wmma_f32_16x16x4_f32(
               false, a[kc], false, b1, (short)0, c1, false, false);
    }
    const int   na  = n0 + lr;
    const int   nb  = na + 16;
    const float ea2 = hn[na];
    const float eb2 = hn[nb];
    #pragma unroll
    for (int r = 0; r < 8; ++r) {
      float va = ea2 - c0[r];                // c: VGPR r -> row (half? 8+r : r), col = lane%16
      if (va < best[r]) { best[r] = va; bidx[r] = na; }
      float vb = eb2 - c1[r];
      if (vb < best[r]) { best[r] = vb; bidx[r] = nb; }
    }
  }

  // Min-reduce (val, idx) across the 16 lanes of each half; first-index tie-break.
  #pragma unroll
  for (int r = 0; r < 8; ++r) {
    #pragma unroll
    for (int m = 1; m < 16; m <<= 1) {
      float ov = __shfl_xor(best[r], m, 32);
      int   oi = __shfl_xor(bidx[r], m, 32);
      if (ov < best[r] || (ov == best[r] && oi < bidx[r])) { best[r] = ov; bidx[r] = oi; }
    }
  }

  if (lr == 0) {
    #pragma unroll
    for (int r = 0; r < 8; ++r) encIdx[r0 + half * 8 + r] = bidx[r];
  }
}

// ---------------------------------------------------------------------------
// Kernel 2: gather qtised + per-block partial sums of (q - x)^2
// ---------------------------------------------------------------------------
__global__ __launch_bounds__(256) void vq_gather_kernel(const float* __restrict__ X,
                                                        const float* __restrict__ E,
                                                        const int* __restrict__ encIdx,
                                                        float* __restrict__ out,
                                                        float* __restrict__ partials) {
  __shared__ float red[256];
  const size_t gid = (size_t)blockIdx.x * 256 + threadIdx.x;
  const int i = (int)(gid >> 6);
  const int d = (int)(gid & 63);
  const int k = encIdx[i];
  const float q = E[d * KCB + k];   // codebook fully L2-resident (128 KB)
  out[gid] = q;
  const float diff = q - X[gid];
  red[threadIdx.x] = diff * diff;
  __syncthreads();
  #pragma unroll
  for (int s = 128; s > 0; s >>= 1) {
    if ((int)threadIdx.x < s) red[threadIdx.x] += red[threadIdx.x + s];
    __syncthreads();
  }
  if (threadIdx.x == 0) partials[blockIdx.x] = red[0];
}

// ---------------------------------------------------------------------------
// Kernel 3: fold partials -> loss = 1.25 * mean((q-x)^2)   (deterministic tree)
// ---------------------------------------------------------------------------
__global__ __launch_bounds__(256) void vq_loss_kernel(const float* __restrict__ partials,
                                                      int np, float* __restrict__ loss_out) {
  __shared__ float red[256];
  float s = 0.f;
  for (int i = threadIdx.x; i < np; i += 256) s += partials[i];
  red[threadIdx.x] = s;
  __syncthreads();
  #pragma unroll
  for (int t = 128; t > 0; t >>= 1) {
    if ((int)threadIdx.x < t) red[threadIdx.x] += red[threadIdx.x + t];
    __syncthreads();
  }
  if (threadIdx.x == 0) *loss_out = 1.25f * red[0] / (float)NTOT;
}

extern "C" void kernel_launch(void* const* d_in, const int* in_sizes, int n_in,
                              void* d_out, int out_size, void* d_ws, size_t ws_size,
                              hipStream_t stream) {
  const float* X = (const float*)d_in[0];   // [32,64,64,64] f32
  const float* E = (const float*)d_in[1];   // [64,512] f32
  float* out = (float*)d_out;               // qtised (NTOT floats) ++ loss (1 float)

  int*   encIdx   = (int*)d_ws;                       // N_ROWS ints
  float* partials = (float*)(encIdx + N_ROWS);        // NTOT/256 floats

  const size_t ldsBytes = (size_t)(KCB * E_STRIDE + KCB) * sizeof(float);

  vq_argmin_kernel<<<N_ROWS / 128, 256, ldsBytes, stream>>>(X, E, encIdx);
  vq_gather_kernel<<<NTOT / 256, 256, 0, stream>>>(X, E, encIdx, out, partials);
  vq_loss_kernel<<<1, 256, 0, stream>>>(partials, NTOT / 256, out + NTOT);
}